// AdditiveNoise_70076686401817
// MI455X (gfx1250) — compile-verified
//
#include <hip/hip_runtime.h>
#include <math.h>

// ---------------------------------------------------------------------------
// AdditiveNoise (pink-noise IIR + SNR scaling) for MI455X / gfx1250, wave32.
//
//   p[i] = 0.02*w[i] + 0.98*p[i-1], p[0]=0
//   out  = audio + sqrt(mean(audio^2)/100) * p
//
// Memory-bound: ~272MB traffic -> ~11.7us @ 23.3TB/s. The serial IIR is
// parallelized exactly (to f32 precision) because 0.98^1024 ~ 1e-9: each
// wave32 reconstructs its initial state from the preceding 1024 samples via
// a geometric-weighted reduction, then scans its 4096-sample span using the
// matrix unit: local 16x16 scan = T(16x16, T[i][j]=0.98^(i-j)) x U done as
// 4 chained v_wmma_f32_16x16x4_f32, with 16-column carries via a shfl
// Kogge-Stone scan (coef 0.98^16) and a scalar batch carry (coef 0.98^256).
// Assumes L % 32768 == 0 (harness uses L = 2^24).
// ---------------------------------------------------------------------------

typedef __attribute__((ext_vector_type(2))) float v2f;
typedef __attribute__((ext_vector_type(8))) float v8f;

#define AN_A 0.98
#define AN_B 0.02f

// 0.98^n via double binary exponentiation (one-time per-thread init cost)
__device__ __forceinline__ float an_apow(int n) {
  double r = 1.0, b = AN_A;
  unsigned e = (unsigned)n;
  while (e) { if (e & 1u) r *= b; b *= b; e >>= 1u; }
  return (float)r;
}

// ---------------- K1: per-block partial sum of squares ---------------------
__global__ __launch_bounds__(256) void an_sumsq_kernel(
    const float* __restrict__ audio, float* __restrict__ partial, int n4) {
  int tid = blockIdx.x * blockDim.x + threadIdx.x;
  int stride = gridDim.x * blockDim.x;
  const float4* a4 = (const float4*)audio;
  float acc = 0.f;
  for (int i = tid; i < n4; i += stride) {
    float4 v = a4[i];
    acc += v.x * v.x + v.y * v.y + v.z * v.z + v.w * v.w;
  }
  #pragma unroll
  for (int off = 16; off; off >>= 1) acc += __shfl_xor(acc, off);
  __shared__ float red[8];
  int lane = threadIdx.x & 31, wid = threadIdx.x >> 5;
  if (lane == 0) red[wid] = acc;
  __syncthreads();
  if (threadIdx.x == 0) {
    float s = 0.f;
    #pragma unroll
    for (int i = 0; i < 8; ++i) s += red[i];
    partial[blockIdx.x] = s;
  }
}

// ---------------- K2: finalize amplitude = sqrt(mean/100) ------------------
__global__ __launch_bounds__(256) void an_amp_kernel(
    const float* __restrict__ partial, int np, float* __restrict__ amp_out,
    float inv_n) {
  float acc = 0.f;
  for (int i = threadIdx.x; i < np; i += blockDim.x) acc += partial[i];
  #pragma unroll
  for (int off = 16; off; off >>= 1) acc += __shfl_xor(acc, off);
  __shared__ float red[8];
  int lane = threadIdx.x & 31, wid = threadIdx.x >> 5;
  if (lane == 0) red[wid] = acc;
  __syncthreads();
  if (threadIdx.x == 0) {
    float s = 0.f;
    #pragma unroll
    for (int i = 0; i < 8; ++i) s += red[i];
    float power = s * inv_n;              // mean(audio^2)
    amp_out[0] = sqrtf(power * 0.01f);    // sqrt(power / 10^(SNR/10)), SNR=20
  }
}

// ---------------- K3: WMMA-based IIR scan + output -------------------------
// Each wave32 owns 4096 samples = 16 batches of 256 (16 segments x 16).
__global__ __launch_bounds__(256) void an_pink_wmma_kernel(
    const float* __restrict__ audio, const float* __restrict__ white,
    const float* __restrict__ ampp, float* __restrict__ out, int n) {
  const int lane = threadIdx.x & 31;
  const int wid  = threadIdx.x >> 5;
  const int m    = lane & 15;   // segment/column index, also row-in-half
  const int h    = lane >> 4;   // half-wave (0: rows 0-7 / K 0-1, 1: rows 8-15 / K 2-3)
  const long base = ((long)blockIdx.x * 8 + wid) * 4096;
  if (base >= (long)n) return;

  const float amp = ampp[0];

  // A-matrix: T[i][j] = 0.98^(i-j) (i>=j), 16x4 column blocks per WMMA step.
  // Layout: lane<16 holds row m, K = {4k, 4k+1}; lane>=16 holds row m, K = {4k+2, 4k+3}.
  v2f Aop[4];
  #pragma unroll
  for (int k = 0; k < 4; ++k) {
    int j0 = 4 * k + 2 * h;
    Aop[k][0] = (m >= j0)     ? an_apow(m - j0)     : 0.f;
    Aop[k][1] = (m >= j0 + 1) ? an_apow(m - j0 - 1) : 0.f;
  }
  // carry-application powers: element (row = r + 8h) gets + a^(row+1) * S_col
  float pwoff[8];
  #pragma unroll
  for (int r = 0; r < 8; ++r) pwoff[r] = an_apow(r + 1 + 8 * h);
  const float a16c  = an_apow(16 * m);   // column decay of batch-initial state
  const float A16_1 = an_apow(16),  A16_2 = an_apow(32);
  const float A16_4 = an_apow(64),  A16_8 = an_apow(128);
  const float A256  = an_apow(256);

  // ---- warm-up: state just before `base` = sum_{k=1..1024} a^(k-1)*B*w[base-k]
  float sigma = 0.f;
  if (base > 0) {                         // base is a multiple of 4096 -> in range
    const int W = 1024;
    const float* wp = white + base - W;
    float wgt = an_apow(W - 1 - lane * 4);
    const float ia1 = 1.f / (float)AN_A;
    const float ia2 = ia1 * ia1, ia3 = ia2 * ia1;
    const float grow = 1.f / an_apow(128);
    float acc = 0.f;
    #pragma unroll
    for (int e = 0; e < W / 128; ++e) {
      float4 v = *(const float4*)(wp + e * 128 + lane * 4);
      acc += wgt * (v.x + ia1 * v.y + ia2 * v.z + ia3 * v.w);
      wgt *= grow;
    }
    #pragma unroll
    for (int off = 16; off; off >>= 1) acc += __shfl_xor(acc, off);
    sigma = AN_B * acc;
  }

  // ---- 16 batches of 256 samples, fully streaming
  for (int j = 0; j < 16; ++j) {
    const long ubase = base + (long)j * 256;

    // D = T * U  (U[i][c] = 0.02*w[ubase + c*16 + i]); 4 chained 16x16x4 WMMAs.
    v8f c = {};
    #pragma unroll
    for (int k = 0; k < 4; ++k) {
      float4 v = *(const float4*)(white + ubase + m * 16 + k * 4);
      float bx = AN_B * v.x, by = AN_B * v.y, bz = AN_B * v.z, bw = AN_B * v.w;
      if (ubase == 0 && m == 0 && k == 0) bx = 0.f;   // u[0] = 0
      v2f Bop;
      Bop[0] = h ? bz : bx;   // B 4x16: lane<16 rows K={4k,4k+1}, lane>=16 K={4k+2,4k+3}
      Bop[1] = h ? bw : by;
      c = __builtin_amdgcn_wmma_f32_16x16x4_f32(false, Aop[k], false, Bop,
                                                (short)0, c, false, false);
    }

    // Column carries: y_c = D[15][c] (lanes 16..31, reg 7).
    float yv = __shfl(c[7], 16 + m);
    // Inclusive Kogge-Stone scan with coefficient a^16 within 16-lane groups.
    float Z = yv, t;
    t = __shfl_up(Z, 1, 16); if (m >= 1) Z += A16_1 * t;
    t = __shfl_up(Z, 2, 16); if (m >= 2) Z += A16_2 * t;
    t = __shfl_up(Z, 4, 16); if (m >= 4) Z += A16_4 * t;
    t = __shfl_up(Z, 8, 16); if (m >= 8) Z += A16_8 * t;
    // State entering column c: exclusive scan + decayed batch-initial state.
    float S = __shfl_up(Z, 1, 16); if (m == 0) S = 0.f;
    S += a16c * sigma;

    // Apply carries, scale, add audio, store (2 x b128 per lane).
    const long addr = ubase + (long)m * 16 + 8 * h;
    float4 a0 = *(const float4*)(audio + addr);
    float4 a1 = *(const float4*)(audio + addr + 4);
    float4 o0, o1;
    o0.x = a0.x + amp * (c[0] + pwoff[0] * S);
    o0.y = a0.y + amp * (c[1] + pwoff[1] * S);
    o0.z = a0.z + amp * (c[2] + pwoff[2] * S);
    o0.w = a0.w + amp * (c[3] + pwoff[3] * S);
    o1.x = a1.x + amp * (c[4] + pwoff[4] * S);
    o1.y = a1.y + amp * (c[5] + pwoff[5] * S);
    o1.z = a1.z + amp * (c[6] + pwoff[6] * S);
    o1.w = a1.w + amp * (c[7] + pwoff[7] * S);
    *(float4*)(out + addr)     = o0;
    *(float4*)(out + addr + 4) = o1;

    // Batch carry: state after 256 samples.
    sigma = A256 * sigma + __shfl(Z, 15, 16);
  }
}

// ---------------------------------------------------------------------------
extern "C" void kernel_launch(void* const* d_in, const int* in_sizes, int n_in,
                              void* d_out, int out_size, void* d_ws, size_t ws_size,
                              hipStream_t stream) {
  const float* audio = (const float*)d_in[0];
  const float* white = (const float*)d_in[1];
  float* out = (float*)d_out;
  const int n = in_sizes[0];            // 2^24

  float* wsf = (float*)d_ws;
  float* partial = wsf;                 // 1024 block partials
  float* amp = wsf + 1024;              // 1 scalar

  const int NB1 = 1024;
  an_sumsq_kernel<<<NB1, 256, 0, stream>>>(audio, partial, n / 4);
  an_amp_kernel<<<1, 256, 0, stream>>>(partial, NB1, amp, 1.0f / (float)n);

  int nb3 = n / 32768;                  // 8 waves x 4096 samples per block
  if (nb3 < 1) nb3 = 1;
  an_pink_wmma_kernel<<<nb3, 256, 0, stream>>>(audio, white, amp, out, n);
}